// Proxy_Anchor_90933047591179
// MI455X (gfx1250) — compile-verified
//
#include <hip/hip_runtime.h>
#include <math.h>

#define NB_CLASSES 20000
#define SZ_EMBED   512
#define MRG        0.1f
#define ALPHA      32.0f
#define BATCH      128

#define CLS_PER_BLK  8
#define PROX_PER_BLK 80   // 8 classes * 10 proxies = 5 WMMA column tiles of 16
#define NTILES       5

typedef __attribute__((ext_vector_type(2))) float v2f;
typedef __attribute__((ext_vector_type(8))) float v8f;

// ---------------- Kernel 1: row-wise L2 normalize X -> Xn ----------------
__global__ void __launch_bounds__(128)
normalize_x_kernel(const float* __restrict__ X, float* __restrict__ Xn) {
  const int row = blockIdx.x;   // 0..127
  const int t   = threadIdx.x;  // 0..127, one float4 each (512 floats/row)
  float4 v = ((const float4*)(X + (size_t)row * SZ_EMBED))[t];
  float ss = v.x*v.x + v.y*v.y + v.z*v.z + v.w*v.w;
  for (int off = 16; off > 0; off >>= 1) ss += __shfl_xor(ss, off, 32);
  __shared__ float wsum[4];
  __shared__ float sinv;
  if ((t & 31) == 0) wsum[t >> 5] = ss;
  __syncthreads();
  if (t == 0) sinv = rsqrtf(wsum[0] + wsum[1] + wsum[2] + wsum[3] + 1e-12f);
  __syncthreads();
  const float inv = sinv;
  float4 o; o.x = v.x*inv; o.y = v.y*inv; o.z = v.z*inv; o.w = v.w*inv;
  ((float4*)(Xn + (size_t)row * SZ_EMBED))[t] = o;
}

// ---------------- Kernel 2: WMMA GEMM + fused proxy-norm + epilogue ------
// Block = 256 threads = 8 waves; wave w owns batch rows [16w, 16w+16).
// Block b owns proxies [80b, 80b+80) == classes [8b, 8b+8).
__global__ void __launch_bounds__(256)
proxy_anchor_main(const float* __restrict__ Xn,
                  const float* __restrict__ proxies,
                  const int*   __restrict__ T,
                  float* __restrict__ logP,
                  float* __restrict__ logN,
                  float* __restrict__ validf) {
  __shared__ int   Tlds[BATCH];
  __shared__ float Psum[CLS_PER_BLK];
  __shared__ float Nsum[CLS_PER_BLK];
  __shared__ int   matched[CLS_PER_BLK];

  const int tid  = threadIdx.x;
  const int w    = tid >> 5;    // wave id -> M tile
  const int lane = tid & 31;
  const int ln   = lane & 15;   // row (A) / column (B,C,D) within tile
  const int h    = lane >> 4;   // lane half -> K pair offset (A/B), M+8 (C/D)

  if (tid < BATCH)       Tlds[tid] = T[tid];
  if (tid < CLS_PER_BLK) { Psum[tid] = 0.f; Nsum[tid] = 0.f; matched[tid] = 0; }
  __syncthreads();

  const int pbase = blockIdx.x * PROX_PER_BLK;
  const int cbase = blockIdx.x * CLS_PER_BLK;

  // 32-bit byte offsets against uniform SGPR bases -> saddr+voffset+imm loads.
  // ISA f32 A/B fragment layout: per lane, K pair at (2h, 2h+1); row/col = ln.
  const unsigned rowBytes = SZ_EMBED * sizeof(float);              // 2048
  const char* Xb = (const char*)Xn;
  const char* Pb = (const char*)proxies;
  unsigned aoff = (unsigned)(w * 16 + ln) * rowBytes + (unsigned)h * 8u;
  unsigned boff = (unsigned)(pbase + ln)  * rowBytes + (unsigned)h * 8u;

  v8f acc[NTILES];
  v2f nrmv[NTILES];
  #pragma unroll
  for (int j = 0; j < NTILES; ++j) {
    nrmv[j].x = 0.f; nrmv[j].y = 0.f;
    #pragma unroll
    for (int i = 0; i < 8; ++i) acc[j][i] = 0.f;
  }

  #pragma unroll 4
  for (int k = 0; k < SZ_EMBED; k += 4) {
    v2f a = *(const v2f*)(Xb + aoff);
    #pragma unroll
    for (int j = 0; j < NTILES; ++j) {
      // tile j is a compile-time byte offset (j*16 rows * 2048B) -> inst IOFFSET
      v2f b = *(const v2f*)(Pb + boff + (unsigned)(j * 16) * rowBytes);
      nrmv[j] += b * b;   // v_pk_fma_f32 directly on the loaded pair
      acc[j] = __builtin_amdgcn_wmma_f32_16x16x4_f32(
          false, a, false, b, (short)0, acc[j], false, false);
    }
    aoff += 16u;   // 4 floats forward per K step
    boff += 16u;
  }

  // Epilogue: scale by proxy inv-norm, exp, per-class reduction in LDS.
  #pragma unroll
  for (int j = 0; j < NTILES; ++j) {
    // fold the v2f partial, then combine the two K-halves (lanes ln / ln+16)
    float half = nrmv[j].x + nrmv[j].y;
    float full = half + __shfl_xor(half, 16, 32);
    float invb = rsqrtf(full + 1e-12f);
    int jcol  = j * 16 + ln;         // column within block's 80 proxies
    int cl    = jcol / 10;           // local class 0..7
    int cglob = cbase + cl;
    float negsum = 0.f;
    #pragma unroll
    for (int i = 0; i < 8; ++i) {
      float cosv = acc[j][i] * invb;             // normalized cosine
      int   b    = w * 16 + i + 8 * h;           // C/D layout: M = i + 8h
      if (Tlds[b] == cglob) {
        atomicAdd(&Psum[cl], __expf(-ALPHA * (cosv - MRG)));
        matched[cl] = 1;
      } else {
        negsum += __expf(ALPHA * (cosv + MRG));
      }
    }
    atomicAdd(&Nsum[cl], negsum);
  }
  __syncthreads();

  if (tid < CLS_PER_BLK) {
    int c = cbase + tid;
    logP[c]   = log1pf(Psum[tid]);
    logN[c]   = log1pf(Nsum[tid]);
    validf[c] = matched[tid] ? 1.0f : 0.0f;
  }
}

// ---------------- Kernel 3: deterministic final reduction ----------------
__global__ void __launch_bounds__(256)
finalize_kernel(const float* __restrict__ logP, const float* __restrict__ logN,
                const float* __restrict__ validf, float* __restrict__ out) {
  __shared__ float sP[256], sN[256], sV[256];
  int t = threadIdx.x;
  float p = 0.f, n = 0.f, v = 0.f;
  for (int c = t; c < NB_CLASSES; c += 256) { p += logP[c]; n += logN[c]; v += validf[c]; }
  sP[t] = p; sN[t] = n; sV[t] = v;
  __syncthreads();
  for (int s = 128; s > 0; s >>= 1) {
    if (t < s) { sP[t] += sP[t + s]; sN[t] += sN[t + s]; sV[t] += sV[t + s]; }
    __syncthreads();
  }
  if (t == 0) {
    float nv = sV[0] < 1.f ? 1.f : sV[0];
    out[0] = sP[0] / nv + sN[0] / (float)NB_CLASSES;
  }
}

extern "C" void kernel_launch(void* const* d_in, const int* in_sizes, int n_in,
                              void* d_out, int out_size, void* d_ws, size_t ws_size,
                              hipStream_t stream) {
  const float* X       = (const float*)d_in[0];
  const int*   T       = (const int*)  d_in[1];
  const float* proxies = (const float*)d_in[2];

  float* Xn    = (float*)d_ws;                  // 128*512 floats
  float* logP  = Xn + (size_t)BATCH * SZ_EMBED; // 20000
  float* logN  = logP + NB_CLASSES;             // 20000
  float* valid = logN + NB_CLASSES;             // 20000

  normalize_x_kernel<<<BATCH, 128, 0, stream>>>(X, Xn);
  proxy_anchor_main<<<NB_CLASSES / CLS_PER_BLK, 256, 0, stream>>>(
      Xn, proxies, T, logP, logN, valid);
  finalize_kernel<<<1, 256, 0, stream>>>(logP, logN, valid, (float*)d_out);
}